// Model_36704790512260
// MI455X (gfx1250) — compile-verified
//
#include <hip/hip_runtime.h>
#include <hip/hip_bf16.h>

typedef __attribute__((ext_vector_type(2))) float v2f;
typedef __attribute__((ext_vector_type(8))) float v8f;

#define F_IN  34
#define F_HID 10
#define F_OUT 4

// ---------------------------------------------------------------------------
// 1) degree init: deg[i] = 1.0 (self loop)
// ---------------------------------------------------------------------------
__global__ void k_deg_init(float* __restrict__ deg, int n) {
    int i = blockIdx.x * blockDim.x + threadIdx.x;
    if (i < n) deg[i] = 1.0f;
}

// ---------------------------------------------------------------------------
// 2) degree accumulate over edges (dst row of edge_index, int64)
// ---------------------------------------------------------------------------
__global__ void k_deg_accum(const long long* __restrict__ ei, float* __restrict__ deg,
                            long long E) {
    long long e = (long long)blockIdx.x * blockDim.x + threadIdx.x;
    if (e >= E) return;
    long long d = ei[E + e];   // edge_index[1][e]
    unsafeAtomicAdd(&deg[d], 1.0f);
}

// ---------------------------------------------------------------------------
// 3) dinv[i] = rsqrt(deg[i])   (deg >= 1 always due to self loops)
// ---------------------------------------------------------------------------
__global__ void k_dinv(const float* __restrict__ deg, float* __restrict__ dinv, int n) {
    int i = blockIdx.x * blockDim.x + threadIdx.x;
    if (i < n) dinv[i] = rsqrtf(deg[i]);
}

// ---------------------------------------------------------------------------
// 4) h = x @ W_gcn via fp32 WMMA (V_WMMA_F32_16X16X4_F32)
//    One wave per 16-row tile of x. K=34 swept 4 at a time (9 WMMA ops),
//    K-tail and N>=10 columns masked to zero arithmetically (no divergence).
//    A 16x4 f32: lanes 0-15 -> K={0,1}, lanes 16-31 -> K={2,3}, M=lane%16.
//    B 4x16  f32: same K split, N=lane%16.
//    D 16x16 f32: VGPR r holds M = r + 8*(lane>=16), N = lane%16.
// ---------------------------------------------------------------------------
__global__ void k_gcn_linear_wmma(const float* __restrict__ x,
                                  const float* __restrict__ W,   // [F_IN, F_HID]
                                  float* __restrict__ h,         // [N, F_HID]
                                  int n_nodes) {
    const int lane = threadIdx.x & 31;
    const int wave = threadIdx.x >> 5;
    const int tile = blockIdx.x * (blockDim.x >> 5) + wave;
    const int row0 = tile * 16;
    if (row0 >= n_nodes) return;               // wave-uniform exit

    const int half = lane >> 4;                // 0: lanes 0-15, 1: lanes 16-31
    const int lid  = lane & 15;
    const int arow = row0 + lid;               // A-matrix row for this lane
    const int koff = half * 2;                 // per-lane-group K offset

    const int ncol  = (lid < F_HID) ? lid : 0; // clamped B column
    const float nmask = (lid < F_HID) ? 1.0f : 0.0f;

    v8f acc = {};
    #pragma unroll
    for (int kk = 0; kk < F_IN; kk += 4) {     // 9 iterations -> 9 v_wmma
        const int k0 = kk + koff;
        const int k1 = k0 + 1;
        const float m0 = (k0 < F_IN) ? 1.0f : 0.0f;
        const float m1 = (k1 < F_IN) ? 1.0f : 0.0f;
        const int k0c = (k0 < F_IN) ? k0 : 0;
        const int k1c = (k1 < F_IN) ? k1 : 0;

        v2f a, b;
        a.x = x[(size_t)arow * F_IN + k0c] * m0;
        a.y = x[(size_t)arow * F_IN + k1c] * m1;
        b.x = W[k0c * F_HID + ncol] * (m0 * nmask);
        b.y = W[k1c * F_HID + ncol] * (m1 * nmask);

        acc = __builtin_amdgcn_wmma_f32_16x16x4_f32(
            /*neg_a=*/false, a, /*neg_b=*/false, b,
            /*c_mod=*/(short)0, acc, /*reuse_a=*/false, /*reuse_b=*/false);
    }

    // Store the F_HID valid columns of the 16x16 D tile.
    if (lid < F_HID) {
        const int mbase = row0 + half * 8;
        #pragma unroll
        for (int r = 0; r < 8; ++r) {
            const int row = mbase + r;
            if (row < n_nodes) h[(size_t)row * F_HID + lid] = acc[r];
        }
    }
}

// ---------------------------------------------------------------------------
// 5) hidden[i,f] = h[i,f] * dinv[i]^2 + b_gcn[f]   (self-loop term + bias;
//    also fully overwrites the poisoned output region before atomics)
// ---------------------------------------------------------------------------
__global__ void k_hidden_init(const float* __restrict__ h, const float* __restrict__ dinv,
                              const float* __restrict__ b_gcn, float* __restrict__ hidden,
                              int n_nodes) {
    int idx = blockIdx.x * blockDim.x + threadIdx.x;
    int total = n_nodes * F_HID;
    if (idx >= total) return;
    int i = idx / F_HID;
    int f = idx - i * F_HID;
    float di = dinv[i];
    hidden[idx] = h[idx] * di * di + b_gcn[f];
}

// ---------------------------------------------------------------------------
// 6) edge scatter: hidden[dst] += h[src] * dinv[src] * dinv[dst]
//    One thread per edge, 10 hardware float atomics (resolve in L2; the 8MB
//    hidden array fits easily in the 192MB L2).
// ---------------------------------------------------------------------------
__global__ void k_edge_scatter(const long long* __restrict__ ei,
                               const float* __restrict__ h,
                               const float* __restrict__ dinv,
                               float* __restrict__ hidden, long long E) {
    long long e = (long long)blockIdx.x * blockDim.x + threadIdx.x;
    if (e >= E) return;
    long long s = ei[e];
    long long d = ei[E + e];
    float norm = dinv[s] * dinv[d];
    const float* hs = h + (size_t)s * F_HID;
    float* hd = hidden + (size_t)d * F_HID;
    #pragma unroll
    for (int f = 0; f < F_HID; ++f) {
        unsafeAtomicAdd(&hd[f], hs[f] * norm);
    }
}

// ---------------------------------------------------------------------------
// 7) output = relu(hidden @ W_fc + b_fc)   (10x4 per node — trivially BW bound)
// ---------------------------------------------------------------------------
__global__ void k_fc_relu(const float* __restrict__ hidden,
                          const float* __restrict__ W_fc,  // [F_HID, F_OUT]
                          const float* __restrict__ b_fc,
                          float* __restrict__ out, int n_nodes) {
    int i = blockIdx.x * blockDim.x + threadIdx.x;
    if (i >= n_nodes) return;
    float hb[F_HID];
    #pragma unroll
    for (int k = 0; k < F_HID; ++k) hb[k] = hidden[(size_t)i * F_HID + k];
    #pragma unroll
    for (int f = 0; f < F_OUT; ++f) {
        float acc = b_fc[f];
        #pragma unroll
        for (int k = 0; k < F_HID; ++k) acc += hb[k] * W_fc[k * F_OUT + f];
        out[(size_t)i * F_OUT + f] = acc > 0.0f ? acc : 0.0f;
    }
}

// ---------------------------------------------------------------------------
// launch
// ---------------------------------------------------------------------------
extern "C" void kernel_launch(void* const* d_in, const int* in_sizes, int n_in,
                              void* d_out, int out_size, void* d_ws, size_t ws_size,
                              hipStream_t stream) {
    const float*     x     = (const float*)d_in[0];       // [N, 34]
    const long long* ei    = (const long long*)d_in[1];   // [2, E] int64
    const float*     W_gcn = (const float*)d_in[2];       // [34, 10]
    const float*     b_gcn = (const float*)d_in[3];       // [10]
    const float*     W_fc  = (const float*)d_in[4];       // [10, 4]
    const float*     b_fc  = (const float*)d_in[5];       // [4]

    const int       n_nodes = in_sizes[0] / F_IN;
    const long long E       = (long long)in_sizes[1] / 2;

    // d_out = [hidden (N*10) | output (N*4)]
    float* hidden = (float*)d_out;
    float* output = hidden + (size_t)n_nodes * F_HID;

    // workspace: h [N*10] | deg [N] | dinv [N]  (~9.6 MB)
    float* h    = (float*)d_ws;
    float* deg  = h + (size_t)n_nodes * F_HID;
    float* dinv = deg + n_nodes;

    const int B = 256;

    // degree + normalization
    k_deg_init<<<(n_nodes + B - 1) / B, B, 0, stream>>>(deg, n_nodes);
    k_deg_accum<<<(int)((E + B - 1) / B), B, 0, stream>>>(ei, deg, E);
    k_dinv<<<(n_nodes + B - 1) / B, B, 0, stream>>>(deg, dinv, n_nodes);

    // h = x @ W_gcn via WMMA: one wave per 16-row tile, 8 waves per block
    const int tiles  = (n_nodes + 15) / 16;
    const int wblocks = (tiles + 7) / 8;
    k_gcn_linear_wmma<<<wblocks, B, 0, stream>>>(x, W_gcn, h, n_nodes);

    // hidden = self-loop term + bias, then scatter-add over edges
    k_hidden_init<<<(n_nodes * F_HID + B - 1) / B, B, 0, stream>>>(h, dinv, b_gcn,
                                                                   hidden, n_nodes);
    k_edge_scatter<<<(int)((E + B - 1) / B), B, 0, stream>>>(ei, h, dinv, hidden, E);

    // output = relu(hidden @ W_fc + b_fc)
    k_fc_relu<<<(n_nodes + B - 1) / B, B, 0, stream>>>(hidden, W_fc, b_fc,
                                                       output, n_nodes);
}